// TLSTM_7344394076795
// MI455X (gfx1250) — compile-verified
//
#include <hip/hip_runtime.h>
#include <hip/hip_bf16.h>
#include <math.h>

typedef __attribute__((ext_vector_type(2))) float v2f;
typedef __attribute__((ext_vector_type(8))) float v8f;

#define B_DIM 4096
#define IN_DIM 1024
#define H_DIM 2048
#define BM 64
#define BN 64
#define BK 32
#define LDT 36   // padded LDS row stride (floats): 144B rows -> 16B aligned, conflict-free frag reads

struct TP {
    const float* x;     // input        [B, IN]
    const float* tm;    // time         [B, 1]
    const float* h;     // hidden       [B, H]
    const float* pc;    // prev_cell    [B, H]
    const float* Ww[4]; // gate input weights  [H, IN]   order: i, f, c, og
    const float* Wb[4];
    const float* Uw[4]; // gate hidden weights [H, H]
    const float* Ub[4];
    const float* bw[4]; // extra hidden weights [H, H]
    const float* bb[4];
    const float* Wd;    // W_decomp_w [H, H]
    const float* Wdb;
    const float* bd;    // b_decomp_w [H, H]
    const float* bdb;
    float* out_h;
    float* out_c;
};

__device__ __forceinline__ float sigmoidf_(float x) {
    return 1.0f / (1.0f + __expf(-x));
}

// Accumulate acc{0,1} += A_tile(64 x K) * (W1 [+ W2])^T tile(K x 64) with f32 WMMA.
// Software-pipelined: the next K-tile's global loads are issued right after the
// fill barrier so they overlap the 16 WMMAs + LDS fragment reads of the current
// tile; registers are committed to LDS at the top of the next iteration.
__device__ __forceinline__ void gemm_acc(
    const float* __restrict__ A, int lda,
    const float* __restrict__ W1, const float* __restrict__ W2, int ldw,
    int K, int Mbase, int Nbase,
    float* sA, float* sW,
    int mi, int n0sub, v8f& acc0, v8f& acc1)
{
    const int t     = threadIdx.x;
    const int lane  = t & 31;
    const int frow  = t >> 2;          // 0..63 : row of the 64-row tile this thread fills
    const int fcol  = (t & 3) << 3;    // 0/8/16/24 : 8 contiguous floats per thread
    const int lmod  = lane & 15;
    const int lhalf = lane >> 4;

    const float* pa  = sA + (mi * 16 + lmod) * LDT + 2 * lhalf;      // A frag: lane=M, vgpr=K%2, half=K>=2
    const float* pw0 = sW + (n0sub * 16 + lmod) * LDT + 2 * lhalf;   // B frag: lane=N (W row = output col)
    const float* pw1 = pw0 + 16 * LDT;

    const float* gA  = A  + (size_t)(Mbase + frow) * lda + fcol;
    const float* gW1 = W1 + (size_t)(Nbase + frow) * ldw + fcol;
    const float* gW2 = W2 ? (W2 + (size_t)(Nbase + frow) * ldw + fcol) : nullptr;
    float* dA = sA + frow * LDT + fcol;
    float* dW = sW + frow * LDT + fcol;

    // Prologue: prefetch first K-tile into registers.
    float4 a0 = *(const float4*)(gA);
    float4 a1 = *(const float4*)(gA + 4);
    float4 w0 = *(const float4*)(gW1);
    float4 w1 = *(const float4*)(gW1 + 4);
    float4 u0 = {0.f, 0.f, 0.f, 0.f}, u1 = {0.f, 0.f, 0.f, 0.f};
    if (gW2) {
        u0 = *(const float4*)(gW2);
        u1 = *(const float4*)(gW2 + 4);
    }

    for (int k0 = 0; k0 < K; k0 += BK) {
        // Fuse second weight matrix (U + b, or W_decomp + b_decomp) at commit time.
        float4 ws0 = w0, ws1 = w1;
        if (gW2) {
            ws0.x += u0.x; ws0.y += u0.y; ws0.z += u0.z; ws0.w += u0.w;
            ws1.x += u1.x; ws1.y += u1.y; ws1.z += u1.z; ws1.w += u1.w;
        }
        __syncthreads();   // previous readers done before refill
        *(float4*)dA       = a0;
        *(float4*)(dA + 4) = a1;
        *(float4*)dW       = ws0;
        *(float4*)(dW + 4) = ws1;
        __syncthreads();

        // Issue next tile's global loads; latency hidden behind WMMA compute below.
        const int kn = k0 + BK;
        if (kn < K) {
            a0 = *(const float4*)(gA + kn);
            a1 = *(const float4*)(gA + kn + 4);
            w0 = *(const float4*)(gW1 + kn);
            w1 = *(const float4*)(gW1 + kn + 4);
            if (gW2) {
                u0 = *(const float4*)(gW2 + kn);
                u1 = *(const float4*)(gW2 + kn + 4);
            }
        }

#pragma unroll
        for (int kk = 0; kk < BK; kk += 4) {
            v2f a  = *(const v2f*)(pa  + kk);
            v2f b0 = *(const v2f*)(pw0 + kk);
            v2f b1 = *(const v2f*)(pw1 + kk);
            acc0 = __builtin_amdgcn_wmma_f32_16x16x4_f32(
                false, a, false, b0, (short)0, acc0, false, false);
            acc1 = __builtin_amdgcn_wmma_f32_16x16x4_f32(
                false, a, false, b1, (short)0, acc1, false, false);
        }
    }
}

__global__ __launch_bounds__(256) void tlstm_fused(TP p)
{
    __shared__ float sA[BM * LDT];
    __shared__ float sW[BN * LDT];

    const int Mbase = blockIdx.x * BM;
    const int Nbase = blockIdx.y * BN;
    const int t     = threadIdx.x;
    const int lane  = t & 31;
    const int wave  = t >> 5;
    const int mi    = wave & 3;               // m sub-tile (16 rows)
    const int n0sub = (wave >> 2) << 1;       // first of two n sub-tiles
    const int lmod  = lane & 15;
    const int lhalf = lane >> 4;

    // D-matrix element coords: lane = N, vgpr r -> M = r + 8*(lane>=16)
    const int row0 = Mbase + mi * 16 + 8 * lhalf;   // + r (0..7)
    const int col0 = Nbase + n0sub * 16 + lmod;
    const int col1 = col0 + 16;

    // ---------------- Phase D: C_ST = tanh(pc @ (Wd + bd)^T + (Wdb + bdb)) ----------------
    v8f acc0 = {}, acc1 = {};
    gemm_acc(p.pc, H_DIM, p.Wd, p.bd, H_DIM, H_DIM, Mbase, Nbase, sA, sW, mi, n0sub, acc0, acc1);

    float Tm1[8];
#pragma unroll
    for (int r = 0; r < 8; ++r) {
        float tv = p.tm[row0 + r];
        Tm1[r] = 1.0f / logf(tv + 2.71828182845904523f) - 1.0f;   // T - 1
    }
    const float bd0 = p.Wdb[col0] + p.bdb[col0];
    const float bd1 = p.Wdb[col1] + p.bdb[col1];
    v8f ca0, ca1;   // cell_adj = prev_cell + (T-1) * C_ST
#pragma unroll
    for (int r = 0; r < 8; ++r) {
        size_t i0 = (size_t)(row0 + r) * H_DIM + col0;
        size_t i1 = (size_t)(row0 + r) * H_DIM + col1;
        ca0[r] = p.pc[i0] + Tm1[r] * tanhf(acc0[r] + bd0);
        ca1[r] = p.pc[i1] + Tm1[r] * tanhf(acc1[r] + bd1);
    }

    v8f ig0, ig1;   // holds gate i, later tanh(c)

    // ---------------- Gates: 0=i, 1=f, 2=c(state), 3=og ----------------
#pragma unroll 1
    for (int g = 0; g < 4; ++g) {
        v8f a0 = {}, a1 = {};
        gemm_acc(p.x, IN_DIM, p.Ww[g], nullptr,  IN_DIM, IN_DIM, Mbase, Nbase, sA, sW, mi, n0sub, a0, a1);
        gemm_acc(p.h, H_DIM,  p.Uw[g], p.bw[g],  H_DIM,  H_DIM,  Mbase, Nbase, sA, sW, mi, n0sub, a0, a1);
        const float bs0 = p.Wb[g][col0] + p.Ub[g][col0] + p.bb[g][col0];
        const float bs1 = p.Wb[g][col1] + p.Ub[g][col1] + p.bb[g][col1];
        if (g == 0) {                       // i
#pragma unroll
            for (int r = 0; r < 8; ++r) {
                ig0[r] = sigmoidf_(a0[r] + bs0);
                ig1[r] = sigmoidf_(a1[r] + bs1);
            }
        } else if (g == 1) {                // f: cell_adj *= f
#pragma unroll
            for (int r = 0; r < 8; ++r) {
                ca0[r] *= sigmoidf_(a0[r] + bs0);
                ca1[r] *= sigmoidf_(a1[r] + bs1);
            }
        } else if (g == 2) {                // state: c = f*cell_adj + i*state
#pragma unroll
            for (int r = 0; r < 8; ++r) {
                float c0 = ca0[r] + ig0[r] * tanhf(a0[r] + bs0);
                float c1 = ca1[r] + ig1[r] * tanhf(a1[r] + bs1);
                p.out_c[(size_t)(row0 + r) * H_DIM + col0] = c0;
                p.out_c[(size_t)(row0 + r) * H_DIM + col1] = c1;
                ig0[r] = tanhf(c0);
                ig1[r] = tanhf(c1);
            }
        } else {                            // og: new_hidden = og * tanh(c)
#pragma unroll
            for (int r = 0; r < 8; ++r) {
                p.out_h[(size_t)(row0 + r) * H_DIM + col0] = sigmoidf_(a0[r] + bs0) * ig0[r];
                p.out_h[(size_t)(row0 + r) * H_DIM + col1] = sigmoidf_(a1[r] + bs1) * ig1[r];
            }
        }
    }
}

extern "C" void kernel_launch(void* const* d_in, const int* in_sizes, int n_in,
                              void* d_out, int out_size, void* d_ws, size_t ws_size,
                              hipStream_t stream) {
    (void)in_sizes; (void)n_in; (void)out_size; (void)d_ws; (void)ws_size;
    TP p;
    p.x  = (const float*)d_in[0];
    p.tm = (const float*)d_in[1];
    p.h  = (const float*)d_in[2];
    p.pc = (const float*)d_in[3];
    // d_in[4]=labels, d_in[5]=keep_prob : unused in forward math
    // params order in d_in: i(6..11) f(12..17) og(18..23) c(24..29) decomp(30..33)
    // our gate loop order: 0=i, 1=f, 2=c, 3=og
    const int base[4] = {6, 12, 24, 18};
    for (int g = 0; g < 4; ++g) {
        p.Ww[g] = (const float*)d_in[base[g] + 0];
        p.Wb[g] = (const float*)d_in[base[g] + 1];
        p.Uw[g] = (const float*)d_in[base[g] + 2];
        p.Ub[g] = (const float*)d_in[base[g] + 3];
        p.bw[g] = (const float*)d_in[base[g] + 4];
        p.bb[g] = (const float*)d_in[base[g] + 5];
    }
    p.Wd  = (const float*)d_in[30];
    p.Wdb = (const float*)d_in[31];
    p.bd  = (const float*)d_in[32];
    p.bdb = (const float*)d_in[33];
    p.out_h = (float*)d_out;
    p.out_c = (float*)d_out + (size_t)B_DIM * H_DIM;

    dim3 grid(B_DIM / BM, H_DIM / BN);
    tlstm_fused<<<grid, 256, 0, stream>>>(p);
}